// SelfAttention_18760417149004
// MI455X (gfx1250) — compile-verified
//
#include <hip/hip_runtime.h>
#include <math.h>

// Problem constants (match reference)
constexpr int NB = 256;   // N
constexpr int L  = 128;
constexpr int E  = 1024;
constexpr int H  = 16;
constexpr int D  = 64;    // E/H
constexpr int S  = 2;

typedef float v2f __attribute__((ext_vector_type(2)));
typedef float v8f __attribute__((ext_vector_type(8)));

__device__ __forceinline__ v8f wmma_f32(v2f a, v2f b, v8f c) {
  // V_WMMA_F32_16X16X4_F32 : D = A(16x4) x B(4x16) + C(16x16)
  return __builtin_amdgcn_wmma_f32_16x16x4_f32(
      /*neg_a=*/false, a, /*neg_b=*/false, b,
      /*c_mod=*/(short)0, c, /*reuse_a=*/false, /*reuse_b=*/false);
}

// ---------------------------------------------------------------------------
// Generic GEMM: C[M x Nw] = gather(X) @ W^T + bias
//   X row r -> gather ? X + (gather[r/Lrows]*Lrows + r%Lrows)*K : X + r*K
//   W: (Nw x K) row-major  =>  B[k][n] = W[n*K + k]
// Block: 128 threads = 4 waves, block tile 32x32, K staged by 32 through LDS.
// ---------------------------------------------------------------------------
__global__ __launch_bounds__(128)
void gemm_xwt_bias(const float* __restrict__ X, const int* __restrict__ gather,
                   int Lrows, const float* __restrict__ W,
                   const float* __restrict__ bias, float* __restrict__ C,
                   int M, int Nw, int K)
{
  __shared__ float As[32][33];
  __shared__ float Bs[32][33];

  const int tid  = threadIdx.x;
  const int lane = tid & 31;
  const int wave = tid >> 5;           // 0..3
  const int tilesN = Nw >> 5;
  const int bm = (int)blockIdx.x / tilesN;   // 32-row block
  const int bn = (int)blockIdx.x % tilesN;   // 32-col block
  const int wr = (wave >> 1) << 4;     // 0 or 16
  const int wc = (wave & 1) << 4;      // 0 or 16

  v8f acc = {0.f,0.f,0.f,0.f,0.f,0.f,0.f,0.f};

  for (int k0 = 0; k0 < K; k0 += 32) {
    // Stage A tile (32 rows x 32 k) — coalesced along k
    #pragma unroll
    for (int i = 0; i < 8; ++i) {
      int idx = tid + i * 128;
      int r = idx >> 5, c = idx & 31;
      int gr = bm * 32 + r;
      long base = gather
          ? ((long)gather[gr / Lrows] * Lrows + (gr % Lrows)) * (long)K
          : (long)gr * K;
      As[r][c] = X[base + k0 + c];
    }
    // Stage B tile: Bs[k][n] = W[(bn*32+n)*K + k0+k] — coalesced along k,
    // padded LDS kills store conflicts.
    #pragma unroll
    for (int i = 0; i < 8; ++i) {
      int idx = tid + i * 128;
      int n = idx >> 5, k = idx & 31;
      Bs[k][n] = W[(long)(bn * 32 + n) * K + k0 + k];
    }
    __syncthreads();

    #pragma unroll
    for (int kk = 0; kk < 32; kk += 4) {
      const int m  = lane & 15;
      const int kb = kk + ((lane >> 4) << 1);
      v2f a, b;
      a.x = As[wr + m][kb];
      a.y = As[wr + m][kb + 1];
      b.x = Bs[kb][wc + m];
      b.y = Bs[kb + 1][wc + m];
      acc = wmma_f32(a, b, acc);
    }
    __syncthreads();
  }

  // C/D layout: vgpr i -> row (lane>=16 ? 8 : 0)+i, col = lane&15
  const int n  = wc + (lane & 15);
  const int mb = wr + ((lane >> 4) << 3);
  const float bval = bias ? bias[bn * 32 + n] : 0.f;
  #pragma unroll
  for (int i = 0; i < 8; ++i) {
    C[(long)(bm * 32 + mb + i) * Nw + bn * 32 + n] = acc[i] + bval;
  }
}

// ---------------------------------------------------------------------------
// energy[s,h,q,k] = (mask==0 ? -1e20 : sum_d Q[q,h,d]*K[s,k,h,d]) * (1/32)
// one wave per 16x16 tile; grid = (64 tiles/4 waves = 16 blocks of 4, S*H)
// ---------------------------------------------------------------------------
__global__ __launch_bounds__(128)
void qk_energy(const float* __restrict__ Qb, const float* __restrict__ Kb,
               const float* __restrict__ mask, float* __restrict__ Emat)
{
  const int lane = threadIdx.x & 31;
  const int wave = threadIdx.x >> 5;
  const int sh = blockIdx.y;           // 0..S*H-1
  const int s = sh / H, h = sh % H;
  const int t  = (int)blockIdx.x * 4 + wave;   // 0..63
  const int tm = (t >> 3) << 4;        // q-tile base
  const int tn = (t & 7) << 4;         // k-tile base
  const int hb = h * D;

  v8f acc = {0.f,0.f,0.f,0.f,0.f,0.f,0.f,0.f};
  const int m = lane & 15;

  #pragma unroll
  for (int kk = 0; kk < D; kk += 4) {
    const int kb = kk + ((lane >> 4) << 1);
    v2f a, b;
    a.x = Qb[(long)(tm + m) * E + hb + kb];
    a.y = Qb[(long)(tm + m) * E + hb + kb + 1];
    b.x = Kb[(long)(s * L + tn + m) * E + hb + kb];
    b.y = Kb[(long)(s * L + tn + m) * E + hb + kb + 1];
    acc = wmma_f32(a, b, acc);
  }

  const int n  = tn + (lane & 15);
  const int mb = tm + ((lane >> 4) << 3);
  #pragma unroll
  for (int i = 0; i < 8; ++i) {
    const int q = mb + i;
    float e = acc[i];
    e = (mask[q * L + n] == 0.f) ? -1e20f : e;
    Emat[((long)sh * L + q) * L + n] = e * 0.03125f;  // 1/sqrt(E) = 1/32
  }
}

// ---------------------------------------------------------------------------
// Row softmax over 128-wide rows; one wave32 per row, 4 elems/lane.
// ---------------------------------------------------------------------------
__global__ __launch_bounds__(256)
void softmax_rows(float* __restrict__ A, int rows)
{
  const int wavesPerBlk = blockDim.x >> 5;
  const int row = (int)blockIdx.x * wavesPerBlk + (int)(threadIdx.x >> 5);
  if (row >= rows) return;
  const int lane = threadIdx.x & 31;
  float* p = A + (long)row * L;

  float v[4];
  float mx = -INFINITY;
  #pragma unroll
  for (int i = 0; i < 4; ++i) { v[i] = p[lane + i * 32]; mx = fmaxf(mx, v[i]); }
  #pragma unroll
  for (int off = 16; off; off >>= 1) mx = fmaxf(mx, __shfl_xor(mx, off, 32));
  float sum = 0.f;
  #pragma unroll
  for (int i = 0; i < 4; ++i) { v[i] = __expf(v[i] - mx); sum += v[i]; }
  #pragma unroll
  for (int off = 16; off; off >>= 1) sum += __shfl_xor(sum, off, 32);
  const float inv = 1.f / sum;
  #pragma unroll
  for (int i = 0; i < 4; ++i) p[lane + i * 32] = v[i] * inv;
}

// ---------------------------------------------------------------------------
// wv[s*L+q, h*D+d] = sum_l attn[s,h,q,l] * V[s*L+l, h*D+d]
// one wave per 16x16 tile; per (s,h): 8 (q) x 4 (d) = 32 tiles -> 8 blocks of 4
// ---------------------------------------------------------------------------
__global__ __launch_bounds__(128)
void attn_v(const float* __restrict__ At, const float* __restrict__ Vb,
            float* __restrict__ WVb)
{
  const int lane = threadIdx.x & 31;
  const int wave = threadIdx.x >> 5;
  const int sh = blockIdx.y;
  const int s = sh / H, h = sh % H;
  const int t  = (int)blockIdx.x * 4 + wave;   // 0..31
  const int tm = (t >> 2) << 4;        // q-tile base (0..112)
  const int tn = (t & 3) << 4;         // d-tile base (0..48)
  const int hb = h * D;

  v8f acc = {0.f,0.f,0.f,0.f,0.f,0.f,0.f,0.f};
  const int m = lane & 15;

  #pragma unroll
  for (int kk = 0; kk < L; kk += 4) {
    const int kb = kk + ((lane >> 4) << 1);
    v2f a, b;
    a.x = At[((long)sh * L + tm + m) * L + kb];
    a.y = At[((long)sh * L + tm + m) * L + kb + 1];
    b.x = Vb[(long)(s * L + kb) * E + hb + tn + m];
    b.y = Vb[(long)(s * L + kb + 1) * E + hb + tn + m];
    acc = wmma_f32(a, b, acc);
  }

  const int n  = tn + (lane & 15);
  const int mb = tm + ((lane >> 4) << 3);
  #pragma unroll
  for (int i = 0; i < 8; ++i) {
    WVb[(long)(s * L + mb + i) * E + hb + n] = acc[i];
  }
}

// ---------------------------------------------------------------------------
extern "C" void kernel_launch(void* const* d_in, const int* in_sizes, int n_in,
                              void* d_out, int out_size, void* d_ws, size_t ws_size,
                              hipStream_t stream)
{
  (void)in_sizes; (void)n_in; (void)out_size; (void)ws_size;
  const float* values = (const float*)d_in[0];
  const float* keys   = (const float*)d_in[1];
  const float* query  = (const float*)d_in[2];
  const float* mask   = (const float*)d_in[3];
  const float* Wv     = (const float*)d_in[4];
  const float* bv     = (const float*)d_in[5];
  const float* Wk     = (const float*)d_in[6];
  const float* bk     = (const float*)d_in[7];
  const float* Wq     = (const float*)d_in[8];
  const float* bq     = (const float*)d_in[9];
  const float* Wo     = (const float*)d_in[10];
  const float* bo     = (const float*)d_in[11];
  const int* token_index = (const int*)d_in[12];
  const int* stride_idx  = (const int*)d_in[13];
  float* out = (float*)d_out;

  float* Qb  = (float*)d_ws;              // L*E      = 131072
  float* Kb  = Qb  + (long)L * E;         // S*L*E    = 262144
  float* Vb  = Kb  + (long)S * L * E;     // S*L*E    = 262144
  float* At  = Vb  + (long)S * L * E;     // S*H*L*L  = 524288
  float* WVb = At  + (long)S * H * L * L; // S*L*E    = 262144  (≈5.5 MB total)

  // 1) Projections (only the gathered batches are ever needed)
  gemm_xwt_bias<<<dim3((L / 32) * (E / 32)), 128, 0, stream>>>(
      query, token_index, L, Wq, bq, Qb, L, E, E);
  gemm_xwt_bias<<<dim3((S * L / 32) * (E / 32)), 128, 0, stream>>>(
      keys, stride_idx, L, Wk, bk, Kb, S * L, E, E);
  gemm_xwt_bias<<<dim3((S * L / 32) * (E / 32)), 128, 0, stream>>>(
      values, stride_idx, L, Wv, bv, Vb, S * L, E, E);

  // 2) energy = QK^T with mask + 1/sqrt(E) scale
  qk_energy<<<dim3(16, S * H), 128, 0, stream>>>(Qb, Kb, mask, At);

  // 3) row softmax over (S*H*L) rows of length L
  softmax_rows<<<dim3((S * H * L) / 8), 256, 0, stream>>>(At, S * H * L);

  // 4) wv = attn @ V, laid out directly as (S*L, E)
  attn_v<<<dim3(8, S * H), 128, 0, stream>>>(At, Vb, WVb);

  // 5) out = wv @ Wo^T + bo  -> (256, 1024)
  gemm_xwt_bias<<<dim3((NB / 32) * (E / 32)), 128, 0, stream>>>(
      WVb, (const int*)nullptr, NB, Wo, bo, out, NB, E, E);
}